// Con_Shallow_19250043420784
// MI455X (gfx1250) — compile-verified
//
#include <hip/hip_runtime.h>
#include <hip/hip_bf16.h>

// Problem constants (match reference)
#define B_  8
#define C_  128
#define H_  160
#define W_  160
#define K_  41          // NUM_CLASSES
#define KP_ 48          // padded to 3x16 WMMA tiles
#define HW_ (H_ * W_)               // 25600
#define CHW_ (C_ * H_ * W_)         // 3276800
#define R_  (B_ * H_ * W_)          // 204800 chunks of C_ elements
#define GEMM_BLOCKS 200
#define SLAB_ (R_ / GEMM_BLOCKS)    // 1024, multiple of 4

typedef __attribute__((ext_vector_type(2))) float v2f;
typedef __attribute__((ext_vector_type(8))) float v8f;

// ---------------- zero scratch ----------------
__global__ void zero_words(int* __restrict__ p, int n) {
    int i = blockIdx.x * blockDim.x + threadIdx.x;
    if (i < n) p[i] = 0;
}

// ---------------- kernel 1: build per-class pooled vectors ----------------
// One thread owns one chunk of 128 consecutive flat elements -> exclusive
// accumulation into a private 41-entry LDS column (no atomics).
__global__ void build_v(const float* __restrict__ feat,
                        const int* __restrict__ labels,
                        float* __restrict__ v /* [KP_][R_] row-major */) {
    __shared__ float acc[K_ * 256];
    const int tid = threadIdx.x;
    const int chunk = blockIdx.x * 256 + tid;   // grid sized so chunk < R_

    #pragma unroll
    for (int cl = 0; cl < K_; ++cl) acc[cl * 256 + tid] = 0.0f;

    const float4* p = (const float4*)(feat + (size_t)chunk * C_);
    const unsigned base = (unsigned)chunk * (unsigned)C_;

    #pragma unroll 4
    for (int i = 0; i < C_ / 4; ++i) {
        float4 x4 = p[i];
        float xs[4] = {x4.x, x4.y, x4.z, x4.w};
        #pragma unroll
        for (int s = 0; s < 4; ++s) {
            unsigned f  = base + (unsigned)(i * 4 + s);
            unsigned b  = f / (unsigned)CHW_;           // batch index
            unsigned hw = (f - b * (unsigned)CHW_) % (unsigned)HW_; // h*W + w
            int l = labels[b * (unsigned)HW_ + hw];
            acc[l * 256 + tid] += xs[s];
        }
    }

    const float invC = 1.0f / (float)C_;
    #pragma unroll
    for (int cl = 0; cl < KP_; ++cl) {
        float val = (cl < K_) ? acc[cl * 256 + tid] * invC : 0.0f;
        v[(size_t)cl * R_ + chunk] = val;   // coalesced across tid
    }
}

// ---------------- kernel 2: per-row inverse L2 norms ----------------
__global__ void row_inv_norms(const float* __restrict__ vs,
                              const float* __restrict__ vt,
                              float* __restrict__ inv /* [2][KP_] */) {
    const int cl = blockIdx.x;      // 0..K_-1
    const int t  = blockIdx.y;      // 0 = student, 1 = teacher
    const float* row = (t == 0 ? vs : vt) + (size_t)cl * R_;
    float s = 0.0f;
    for (int r = threadIdx.x; r < R_; r += 256) {
        float x = row[r];
        s += x * x;
    }
    __shared__ float red[256];
    red[threadIdx.x] = s;
    __syncthreads();
    for (int o = 128; o > 0; o >>= 1) {
        if (threadIdx.x < o) red[threadIdx.x] += red[threadIdx.x + o];
        __syncthreads();
    }
    if (threadIdx.x == 0) {
        float n = sqrtf(red[0]);
        inv[t * KP_ + cl] = 1.0f / fmaxf(n, 1e-12f);
    }
}

// ---------------- kernel 3: class presence ----------------
__global__ void presence_kernel(const int* __restrict__ labels,
                                int* __restrict__ present) {
    int i = blockIdx.x * blockDim.x + threadIdx.x;
    if (i < R_) atomicOr(&present[labels[i]], 1);
}

// ---------------- kernel 4: 48x48 fp32 GEMM via V_WMMA_F32_16X16X4_F32 ----
// raw[i][j] += sum_k vs[i][k] * vt[j][k] over this block's K-slab.
// 9 waves per block: wave w owns output tile (w/3, w%3).
__global__ void gemm_wmma(const float* __restrict__ vs,
                          const float* __restrict__ vt,
                          float* __restrict__ raw /* [KP_][KP_] */) {
    const int wave = threadIdx.x >> 5;
    const int lane = threadIdx.x & 31;
    const int ti = wave / 3, tj = wave % 3;

    const int mr = lane & 15;            // row within tile for A, col for B
    const int koff = (lane >> 4) * 2;    // lanes 16-31 hold K+2, K+3

    const float* rowA = vs + (size_t)(ti * 16 + mr) * R_;
    const float* rowB = vt + (size_t)(tj * 16 + mr) * R_;

    const int k0 = blockIdx.x * SLAB_;
    v8f c = {0.f, 0.f, 0.f, 0.f, 0.f, 0.f, 0.f, 0.f};

    for (int k = k0; k < k0 + SLAB_; k += 4) {
        v2f a = *(const v2f*)(rowA + k + koff);
        v2f b = *(const v2f*)(rowB + k + koff);
        // D = A(16x4) * B(4x16) + C ; fp32 throughout (matches reference)
        c = __builtin_amdgcn_wmma_f32_16x16x4_f32(
                /*neg_a=*/false, a, /*neg_b=*/false, b,
                /*c_mod=*/(short)0, c, /*reuse_a=*/false, /*reuse_b=*/false);
    }

    // C/D layout: vgpr v -> M = v + 8*(lane/16), N = lane%16
    const int mg0 = ti * 16 + (lane >> 4) * 8;
    const int ng  = tj * 16 + (lane & 15);
    #pragma unroll
    for (int v = 0; v < 8; ++v) {
        atomicAdd(&raw[(mg0 + v) * KP_ + ng], c[v]);
    }
}

// ---------------- kernel 5: logsumexp contrastive loss ----------------
__global__ void finalize_loss(const float* __restrict__ raw,
                              const float* __restrict__ inv,
                              const int* __restrict__ present,
                              float* __restrict__ out) {
    __shared__ float terms[64];
    __shared__ int   pcnt[64];
    const int i = threadIdx.x;
    float term = 0.0f;
    int p = 0;
    if (i < K_) {
        const float si = inv[i];
        // pass 1: max + diagonal
        float mx = -3.4e38f, diag = 0.0f;
        for (int j = 0; j < K_; ++j) {
            float lij = raw[i * KP_ + j] * si * inv[KP_ + j] * 10.0f; // 1/TEMP
            if (j == i) diag = lij;
            mx = fmaxf(mx, lij);
        }
        // pass 2: sum of exp
        float s = 0.0f;
        for (int j = 0; j < K_; ++j) {
            float lij = raw[i * KP_ + j] * si * inv[KP_ + j] * 10.0f;
            s += __expf(lij - mx);
        }
        float lse = mx + __logf(s);
        term = lse - diag;
        p = present[i] ? 1 : 0;
    }
    terms[i] = p ? term : 0.0f;
    pcnt[i] = p;
    __syncthreads();
    for (int o = 32; o > 0; o >>= 1) {
        if (i < o) { terms[i] += terms[i + o]; pcnt[i] += pcnt[i + o]; }
        __syncthreads();
    }
    if (i == 0) out[0] = terms[0] / (float)(pcnt[0] > 0 ? pcnt[0] : 1);
}

extern "C" void kernel_launch(void* const* d_in, const int* in_sizes, int n_in,
                              void* d_out, int out_size, void* d_ws, size_t ws_size,
                              hipStream_t stream) {
    (void)in_sizes; (void)n_in; (void)out_size; (void)ws_size;
    const float* feat_s = (const float*)d_in[0];
    const float* feat_t = (const float*)d_in[1];
    const int*   labels = (const int*)d_in[2];
    float* out = (float*)d_out;

    char* ws = (char*)d_ws;
    const size_t vbytes = (size_t)KP_ * R_ * sizeof(float);   // 39.3 MB
    float* v_s     = (float*)ws;
    float* v_t     = (float*)(ws + vbytes);
    float* raw     = (float*)(ws + 2 * vbytes);               // 48*48
    float* inv     = raw + KP_ * KP_;                         // 2*48
    int*   present = (int*)(inv + 2 * KP_);                   // 41

    // zero raw + inv + present (contiguous words)
    const int zwords = KP_ * KP_ + 2 * KP_ + K_;
    zero_words<<<(zwords + 255) / 256, 256, 0, stream>>>((int*)raw, zwords);

    // per-class pooled vectors (exclusive per-thread chunks, LDS scatter)
    build_v<<<R_ / 256, 256, 0, stream>>>(feat_s, labels, v_s);
    build_v<<<R_ / 256, 256, 0, stream>>>(feat_t, labels, v_t);

    // inverse row norms
    row_inv_norms<<<dim3(K_, 2), 256, 0, stream>>>(v_s, v_t, inv);

    // class presence mask
    presence_kernel<<<(R_ + 255) / 256, 256, 0, stream>>>(labels, present);

    // 48x48 fp32 WMMA GEMM over K = 204800, split into 200 K-slabs
    gemm_wmma<<<GEMM_BLOCKS, 9 * 32, 0, stream>>>(v_s, v_t, raw);

    // scalar loss
    finalize_loss<<<1, 64, 0, stream>>>(raw, inv, present, out);
}